// MipHistogramLossMasked_60576218742870
// MI455X (gfx1250) — compile-verified
//
#include <hip/hip_runtime.h>
#include <hip/hip_bf16.h>
#include <stdint.h>

// Problem constants (match reference: C=256, H=W=256, BINS=256, 3 levels)
#define CH_N   65536      // H*W elements per channel
#define NCH    768        // 3 levels * 256 channels
#define CPL    256        // channels per level
#define NBINS  256
#define TPB    256        // 8 waves of 32 (wave32)

typedef __attribute__((ext_vector_type(2))) float v2f_;
typedef __attribute__((ext_vector_type(8))) float v8f_;

// order-preserving float -> uint32 key (matches jnp sort order; -0 < +0)
__device__ __forceinline__ unsigned fkey(float x) {
  unsigned u = __float_as_uint(x);
  return (u & 0x80000000u) ? ~u : (u | 0x80000000u);
}

// CDNA5 async global->LDS copy (ASYNCcnt-tracked), 16B per lane
__device__ __forceinline__ void async_ld16(unsigned lds_byte_off, const void* gaddr) {
  asm volatile("global_load_async_to_lds_b128 %0, %1, off"
               :: "v"(lds_byte_off), "v"(gaddr)
               : "memory");
}
__device__ __forceinline__ void wait_async0() {
  asm volatile("s_wait_asynccnt 0x0" ::: "memory");
}
__device__ __forceinline__ void wait_async1() {
  asm volatile("s_wait_asynccnt 0x1" ::: "memory");
}

__device__ __forceinline__ unsigned gload(unsigned* p) {
  return __hip_atomic_load(p, __ATOMIC_RELAXED, __HIP_MEMORY_SCOPE_AGENT);
}
__device__ __forceinline__ void gstore(unsigned* p, unsigned v) {
  __hip_atomic_store(p, v, __ATOMIC_RELAXED, __HIP_MEMORY_SCOPE_AGENT);
}

// Reduce 256 floats staged in LDS with a chain of 4 v_wmma_f32_16x16x4_f32
// (A = 16x4 chunk of partials, B = ones => every D column holds row sums;
//  C accumulates across the 4 chunks). All 8 waves compute redundantly so
// EXEC is all-1s without branching; caller uses the value from thread 0.
__device__ __forceinline__ float wmma_reduce256(const float* s_part) {
  int lane = threadIdx.x & 31;
  v8f_ cacc = {};
  v2f_ bones; bones.x = 1.0f; bones.y = 1.0f;
  int m  = lane & 15;
  int k0 = (lane < 16) ? 0 : 2;
#pragma unroll
  for (int t = 0; t < 4; ++t) {
    v2f_ a;
    a.x = s_part[t * 64 + m * 4 + k0];
    a.y = s_part[t * 64 + m * 4 + k0 + 1];
    cacc = __builtin_amdgcn_wmma_f32_16x16x4_f32(false, a, false, bones,
                                                 (short)0, cacc, false, false);
  }
  // lane 0 holds D[M=0..7][0] in cacc[0..7]; lane 16 holds D[M=8..15][0]
  float srow = cacc[0] + cacc[1] + cacc[2] + cacc[3] +
               cacc[4] + cacc[5] + cacc[6] + cacc[7];
  return __shfl(srow, 0, 32) + __shfl(srow, 16, 32);
}

__global__ void __launch_bounds__(TPB)
mip_hist_channel_kernel(const float* __restrict__ opt0, const float* __restrict__ opt1,
                        const float* __restrict__ opt2,
                        const float* __restrict__ hist0, const float* __restrict__ hist1,
                        const float* __restrict__ hist2,
                        const float* __restrict__ minv0, const float* __restrict__ maxv0,
                        const float* __restrict__ minv1, const float* __restrict__ maxv1,
                        const float* __restrict__ minv2, const float* __restrict__ maxv2,
                        const int* __restrict__ mask,
                        float* __restrict__ partial_diff,   // [NCH]
                        float* __restrict__ partial_cnt,    // [CPL]
                        unsigned* __restrict__ slots)       // gridDim.x * 2 * 65536 u32
{
  __shared__ float    s_cdf[NBINS];
  __shared__ float    s_stage[2][TPB * 4];    // async double buffer, 8 KB
  __shared__ unsigned s_sums[TPB];
  __shared__ float    s_part[TPB];

  unsigned* histg   = slots + (size_t)blockIdx.x * (2u * 65536u);
  unsigned* grouped = histg + 65536;
  const int tid = threadIdx.x;

  const unsigned lds0 = (unsigned)(uintptr_t)&s_stage[0][tid * 4];
  const unsigned lds1 = (unsigned)(uintptr_t)&s_stage[1][tid * 4];

  for (int ch = blockIdx.x; ch < NCH; ch += gridDim.x) {
    const int l = ch >> 8, c = ch & 255;
    const float* opt  = (l == 0 ? opt0 : (l == 1 ? opt1 : opt2)) + (size_t)c * CH_N;
    const float* hrow = (l == 0 ? hist0 : (l == 1 ? hist1 : hist2)) + (size_t)c * NBINS;
    const float  lov  = (l == 0 ? minv0 : (l == 1 ? minv1 : minv2))[c];
    const float  hiv  = (l == 0 ? maxv0 : (l == 1 ? maxv1 : maxv2))[c];
    const int*   mrow = mask + (size_t)c * CH_N;

    __syncthreads();
    // serial f32 cumsum to match jnp.cumsum rounding exactly
    if (tid == 0) {
      float acc = 0.0f;
      for (int b = 0; b < NBINS; ++b) { acc += hrow[b]; s_cdf[b] = acc; }
    }
    // zero the 64K-bucket histogram (L2-resident scratch)
    for (int j = tid; j < CH_N; j += TPB) gstore(&histg[j], 0u);
    __syncthreads();
    const float total = s_cdf[NBINS - 1];
    const float range = hiv - lov;

    // ---------------- Phase A: stream via async global->LDS, histogram + masked opt sum
    float accOpt = 0.0f, accCnt = 0.0f;
    const int NIT = CH_N / (TPB * 4);   // 64 tiles of 1024 floats
    async_ld16(lds0, opt + tid * 4);
    for (int it = 0; it < NIT; ++it) {
      const int i0 = it * TPB * 4 + tid * 4;
      if (it + 1 < NIT) {
        async_ld16(((it & 1) ? lds0 : lds1), opt + (it + 1) * TPB * 4 + tid * 4);
        wait_async1();                  // allow the prefetch to stay in flight
      } else {
        wait_async0();
      }
      const float* buf = s_stage[it & 1];
      const int4 mv = *(const int4*)(mrow + i0);
      float x0 = buf[tid * 4 + 0], x1 = buf[tid * 4 + 1];
      float x2 = buf[tid * 4 + 2], x3 = buf[tid * 4 + 3];
      atomicAdd(&histg[fkey(x0) >> 16], 1u);
      atomicAdd(&histg[fkey(x1) >> 16], 1u);
      atomicAdd(&histg[fkey(x2) >> 16], 1u);
      atomicAdd(&histg[fkey(x3) >> 16], 1u);
      if (mv.x) { accOpt += x0; accCnt += 1.0f; }
      if (mv.y) { accOpt += x1; accCnt += 1.0f; }
      if (mv.z) { accOpt += x2; accCnt += 1.0f; }
      if (mv.w) { accOpt += x3; accCnt += 1.0f; }
    }
    __syncthreads();
    __threadfence();

    // ---------------- Phase B: exclusive prefix over 64K buckets
    {
      const unsigned base = (unsigned)tid * 256u;
      unsigned ls = 0;
      for (int j = 0; j < 256; ++j) ls += gload(&histg[base + j]);
      s_sums[tid] = ls;
      __syncthreads();
      if (tid == 0) {
        unsigned run = 0;
        for (int i = 0; i < TPB; ++i) { unsigned v = s_sums[i]; s_sums[i] = run; run += v; }
      }
      __syncthreads();
      unsigned run = s_sums[tid];
      for (int j = 0; j < 256; ++j) {
        unsigned v = gload(&histg[base + j]);
        gstore(&histg[base + j], run);
        run += v;
      }
    }
    __syncthreads();
    __threadfence();

    // ---------------- Phase C: scatter (lo16<<16 | idx) into bucket-grouped array
    for (int it = 0; it < NIT; ++it) {
      const int i0 = it * TPB * 4 + tid * 4;
      const float4 xv = *(const float4*)(opt + i0);
      const float xs[4] = { xv.x, xv.y, xv.z, xv.w };
#pragma unroll
      for (int j = 0; j < 4; ++j) {
        unsigned k = fkey(xs[j]);
        unsigned pos = atomicAdd(&histg[k >> 16], 1u);   // cursor = running prefix
        gstore(&grouped[pos], ((k & 0xFFFFu) << 16) | (unsigned)(i0 + j));
      }
    }
    __syncthreads();
    __threadfence();
    // now histg[h] == inclusive prefix: bucket h spans [histg[h-1], histg[h])

    // ---------------- Phase D: exact rank within bucket -> matched value, masked sum
    float accMatch = 0.0f;
    for (int h = tid; h < CH_N; h += TPB) {
      unsigned s = (h == 0) ? 0u : gload(&histg[h - 1]);
      unsigned e = gload(&histg[h]);
      for (unsigned p = s; p < e; ++p) {
        unsigned me = gload(&grouped[p]);
        unsigned r = s;                            // strict-less count over distinct keys
        for (unsigned q = s; q < e; ++q) r += (gload(&grouped[q]) < me) ? 1u : 0u;
        // t = ((r+1)/N) * total  (f32, matches reference)
        float t = ((float)(r + 1) / 65536.0f) * total;
        // searchsorted(cdf, t, 'left'), clipped to [0, 255]
        int blo = 0, bhi = NBINS;
        while (blo < bhi) { int mid = (blo + bhi) >> 1; if (s_cdf[mid] < t) blo = mid + 1; else bhi = mid; }
        int b = (blo > NBINS - 1) ? (NBINS - 1) : blo;
        float val = ((float)b / 255.0f) * range + lov;
        unsigned idx = me & 0xFFFFu;
        if (mrow[idx] != 0) accMatch += val;
      }
    }

    // ---------------- Block reduction of (opt - matched) partials via WMMA
    __syncthreads();
    s_part[tid] = accOpt - accMatch;
    __syncthreads();
    float diff = wmma_reduce256(s_part);
    if (tid == 0) partial_diff[ch] = diff;

    // mask count (level 0 only), fixed-order serial => deterministic
    __syncthreads();
    s_part[tid] = accCnt;
    __syncthreads();
    if (l == 0 && tid == 0) {
      float sc = 0.0f;
      for (int i = 0; i < TPB; ++i) sc += s_part[i];
      partial_cnt[c] = sc;
    }
  }
}

__global__ void mip_hist_finalize_kernel(const float* __restrict__ partial_diff,
                                         const float* __restrict__ partial_cnt,
                                         const float* __restrict__ w,
                                         float* __restrict__ out) {
  if (blockIdx.x == 0 && threadIdx.x == 0) {
    float cnt = 0.0f;
    for (int i = 0; i < CPL; ++i) cnt += partial_cnt[i];
    float loss = 0.0f;
    for (int l = 0; l < 3; ++l) {
      float S = 0.0f;
      for (int i = 0; i < CPL; ++i) S += partial_diff[l * CPL + i];
      loss += w[l] * (S / cnt);
    }
    out[0] = loss;
  }
}

extern "C" void kernel_launch(void* const* d_in, const int* in_sizes, int n_in,
                              void* d_out, int out_size, void* d_ws, size_t ws_size,
                              hipStream_t stream) {
  (void)in_sizes; (void)n_in; (void)out_size;
  // setup_inputs() dict order: per level i: opt_i, hist_i, minv_i, maxv_i; then
  // mip_weights, mask, bins.
  const float* opt0  = (const float*)d_in[0];
  const float* hist0 = (const float*)d_in[1];
  const float* minv0 = (const float*)d_in[2];
  const float* maxv0 = (const float*)d_in[3];
  const float* opt1  = (const float*)d_in[4];
  const float* hist1 = (const float*)d_in[5];
  const float* minv1 = (const float*)d_in[6];
  const float* maxv1 = (const float*)d_in[7];
  const float* opt2  = (const float*)d_in[8];
  const float* hist2 = (const float*)d_in[9];
  const float* minv2 = (const float*)d_in[10];
  const float* maxv2 = (const float*)d_in[11];
  const float* wgt   = (const float*)d_in[12];
  const int*   mask  = (const int*)d_in[13];

  char* ws = (char*)d_ws;
  float*    partial_diff = (float*)ws;                  // 768 floats
  float*    partial_cnt  = (float*)(ws + NCH * 4);      // 256 floats
  unsigned* slots        = (unsigned*)(ws + 4096);      // 256B-aligned-ish slot area

  const size_t slotBytes = (size_t)2 * 65536 * sizeof(unsigned);   // 512 KB / block
  size_t avail = (ws_size > 4096) ? (ws_size - 4096) : 0;
  int numSlots = (int)(avail / slotBytes);
  if (numSlots < 1)   numSlots = 1;
  if (numSlots > 128) numSlots = 128;   // keep hist+grouped scratch L2-resident (<=64 MB)

  mip_hist_channel_kernel<<<numSlots, TPB, 0, stream>>>(
      opt0, opt1, opt2, hist0, hist1, hist2,
      minv0, maxv0, minv1, maxv1, minv2, maxv2,
      mask, partial_diff, partial_cnt, slots);

  mip_hist_finalize_kernel<<<1, 32, 0, stream>>>(partial_diff, partial_cnt, wgt,
                                                 (float*)d_out);
}